// NS_Block_15728170238690
// MI455X (gfx1250) — compile-verified
//
#include <hip/hip_runtime.h>
#include <hip/hip_bf16.h>

typedef float v2f __attribute__((ext_vector_type(2)));
typedef float v8f __attribute__((ext_vector_type(8)));

constexpr int kB = 8, kT = 8, kC = 32, kH = 16, kW = 28, kHW = kH * kW; // 448
constexpr int kNH = 4, kDK = 8;
constexpr float kLnEps = 1e-5f;
constexpr float kInvScale = 0.35355339059327379f; // 1/sqrt(C/NH) = 1/sqrt(8)

__device__ __forceinline__ v8f wmma_f32_16x16x4(v2f a, v2f b, v8f c) {
    // V_WMMA_F32_16X16X4_F32 : A 16x4 f32, B 4x16 f32, C/D 16x16 f32
    return __builtin_amdgcn_wmma_f32_16x16x4_f32(
        /*neg_a=*/false, a, /*neg_b=*/false, b,
        /*c_mod=*/(short)0, c, /*reuse_a=*/false, /*reuse_b=*/false);
}

// ---------------------------------------------------------------------------
// Intermediate layouts (in d_ws), chosen so the per-pixel 8-channel vectors
// the attention core needs are contiguous 32B:
//   qbuf/kbuf/vbuf/mtbuf : [b][t][pixel][32ch]   (pixel-major)
//   invz                 : [b][t][head][pixel]
// ---------------------------------------------------------------------------

// ---------------------------------------------------------------------------
// Kernel 1: q/k/v projections.  out(32x448) = W(32x32) @ in(32x448) per (b,t).
// One wave per 16x16 output tile, K swept in steps of 4 via WMMA f32.
// D-fragment: lane holds 8 consecutive rows (channels) of one column (pixel)
// -> store as two float4 into the pixel-major buffer (global_store_b128).
// jobs = 64 (b,t) * 3 mats * 2 mtiles * 28 ntiles = 10752; 8 waves/block.
// ---------------------------------------------------------------------------
__global__ void proj_qkv_kernel(const float* __restrict__ first,
                                const float* __restrict__ x,
                                const float* __restrict__ Wq,
                                const float* __restrict__ Wk,
                                const float* __restrict__ Wv,
                                float* __restrict__ qbuf,
                                float* __restrict__ kbuf,
                                float* __restrict__ vbuf) {
    const int lane = threadIdx.x & 31;
    const int wave = threadIdx.x >> 5;
    int job = blockIdx.x * 8 + wave;
    const int ntile = job % 28; job /= 28;
    const int mtile = job % 2;  job /= 2;
    const int mat   = job % 3;  job /= 3;
    const int bt    = job;                      // 0..63

    const float* in = (mat == 0) ? first : x;
    const float* W  = (mat == 0) ? Wq : (mat == 1) ? Wk : Wv;
    float* out      = (mat == 0) ? qbuf : (mat == 1) ? kbuf : vbuf;

    const float* inp = in + (size_t)bt * kC * kHW;   // reference layout [c][p]
    const int m     = lane & 15;
    const int khalf = lane >> 4;                // lanes 16-31 hold K+2,K+3
    const int col   = ntile * 16 + m;

    v8f acc = {0.f, 0.f, 0.f, 0.f, 0.f, 0.f, 0.f, 0.f};
#pragma unroll
    for (int k0 = 0; k0 < kC; k0 += 4) {
        const int c0 = k0 + 2 * khalf;
        v2f a, b;
        a[0] = W[(mtile * 16 + m) * kC + c0];
        a[1] = W[(mtile * 16 + m) * kC + c0 + 1];
        b[0] = inp[(size_t)c0 * kHW + col];
        b[1] = inp[(size_t)(c0 + 1) * kHW + col];
        acc = wmma_f32_16x16x4(a, b, acc);
    }

    // pixel-major store: channels mtile*16 + 8*khalf + (0..7) are contiguous
    float* ob = out + (((size_t)bt * kHW + col) * kC) + mtile * 16 + 8 * khalf;
    float4 lo = make_float4(acc[0], acc[1], acc[2], acc[3]);
    float4 hi = make_float4(acc[4], acc[5], acc[6], acc[7]);
    reinterpret_cast<float4*>(ob)[0] = lo;
    reinterpret_cast<float4*>(ob)[1] = hi;
}

// ---------------------------------------------------------------------------
// Kernel 2: per-(b,t,head) LayerNorm of q over DK*H*W = 3584 elems, in place.
// 448 threads: thread = pixel, owns its contiguous 8 channels (b128 ld/st).
// ---------------------------------------------------------------------------
__global__ void ln_q_kernel(float* __restrict__ qbuf,
                            const float* __restrict__ gamma,
                            const float* __restrict__ beta) {
    const int bt  = blockIdx.x >> 2;            // 0..63
    const int hd  = blockIdx.x & 3;
    const int p   = threadIdx.x;                // 0..447
    const int tid = threadIdx.x;

    float* row = qbuf + (((size_t)bt * kHW + p) * kC) + hd * kDK;
    float v[8];
#pragma unroll
    for (int c = 0; c < 8; ++c) v[c] = row[c];

    float s = 0.f, sq = 0.f;
#pragma unroll
    for (int c = 0; c < 8; ++c) { s += v[c]; sq += v[c] * v[c]; }

    __shared__ float ssum[512];
    __shared__ float ssq[512];
    ssum[tid] = s;
    ssq[tid]  = sq;
    if (tid < 64) { ssum[448 + tid] = 0.f; ssq[448 + tid] = 0.f; }
    __syncthreads();
    for (int off = 256; off > 0; off >>= 1) {
        if (tid < off) {
            ssum[tid] += ssum[tid + off];
            ssq[tid]  += ssq[tid + off];
        }
        __syncthreads();
    }
    const float n    = (float)(kDK * kHW);      // 3584
    const float mu   = ssum[0] / n;
    const float var  = ssq[0] / n - mu * mu;
    const float rstd = rsqrtf(var + kLnEps);

#pragma unroll
    for (int c = 0; c < 8; ++c) {
        row[c] = (v[c] - mu) * rstd * gamma[c * kHW + p] + beta[c * kHW + p];
    }
}

// ---------------------------------------------------------------------------
// Kernel 3: dilated neighborhood attention core.
// Block = (b, head, t); thread = pixel.  Loop s outermost; stage the
// (448 pixel x 8 ch) k and v slices for (b,head,s) into LDS (2 x 14KB),
// then sweep the 49 dilated offsets out of LDS (49x reuse per value).
// Online max/sum-exp -> head max softmax prob = 1/Z.
// Tout uses the *raw scaled score* (per reference: Wm = masked A/scale).
// ---------------------------------------------------------------------------
__global__ void attn_kernel(const float* __restrict__ qbuf,
                            const float* __restrict__ kbuf,
                            const float* __restrict__ vbuf,
                            float* __restrict__ mtbuf,
                            float* __restrict__ invz) {
    __shared__ float4 ks4[kHW * 2];             // [pixel][8ch] as 2 x float4
    __shared__ float4 vs4[kHW * 2];

    const int p = threadIdx.x;                  // 0..447
    const int h = p / kW, w = p % kW;
    int idx = blockIdx.x;
    const int t  = idx & 7;  idx >>= 3;
    const int hd = idx & 3;  idx >>= 2;
    const int b  = idx;                         // 0..7
    const int dil = (hd == 0) ? 1 : (hd == 1) ? 3 : (hd == 2) ? 5 : 7;

    float q[8];
    {
        const float* qr =
            qbuf + (((size_t)(b * kT + t) * kHW + p) * kC) + hd * kDK;
#pragma unroll
        for (int c = 0; c < 8; ++c) q[c] = qr[c];
    }

    float mrun = -3.402823466e38f;
    float Z = 0.f;
    float tout[8];
#pragma unroll
    for (int c = 0; c < 8; ++c) tout[c] = 0.f;

    const float* ksf = reinterpret_cast<const float*>(ks4);
    const float* vsf = reinterpret_cast<const float*>(vs4);

    for (int s = 0; s < kT; ++s) {
        // cooperative stage of k/v slice for (b, s, head): b128 in, b128 to LDS
        const float4* kg = reinterpret_cast<const float4*>(
            kbuf + (((size_t)(b * kT + s) * kHW + p) * kC) + hd * kDK);
        const float4* vg = reinterpret_cast<const float4*>(
            vbuf + (((size_t)(b * kT + s) * kHW + p) * kC) + hd * kDK);
        ks4[p * 2 + 0] = kg[0];
        ks4[p * 2 + 1] = kg[1];
        vs4[p * 2 + 0] = vg[0];
        vs4[p * 2 + 1] = vg[1];
        __syncthreads();

        for (int dy = -3; dy <= 3; ++dy) {
            const int y = h + dy * dil;
            if (y < 0 || y >= kH) continue;
            for (int dx = -3; dx <= 3; ++dx) {
                const int xo = w + dx * dil;
                if (xo < 0 || xo >= kW) continue;
                const int pp = y * kW + xo;
                const float* kk = ksf + pp * 8;
                const float* vv = vsf + pp * 8;
                float sc = 0.f;
#pragma unroll
                for (int c = 0; c < 8; ++c) sc += q[c] * kk[c];
                const float a = sc * kInvScale;
                if (a > mrun) { Z = Z * __expf(mrun - a); mrun = a; }
                Z += __expf(a - mrun);
#pragma unroll
                for (int c = 0; c < 8; ++c) tout[c] += a * vv[c];
            }
        }
        __syncthreads();                        // protect LDS before reload
    }

    // max softmax prob of this head at this pixel = 1/Z  (Z >= 1 always)
    invz[(((size_t)(b * kT + t)) * kNH + hd) * kHW + p] = 1.0f / Z;

    float* ob = mtbuf + (((size_t)(b * kT + t) * kHW + p) * kC) + hd * kDK;
    float4 lo = make_float4(tout[0], tout[1], tout[2], tout[3]);
    float4 hi = make_float4(tout[4], tout[5], tout[6], tout[7]);
    reinterpret_cast<float4*>(ob)[0] = lo;
    reinterpret_cast<float4*>(ob)[1] = hi;
}

// ---------------------------------------------------------------------------
// Kernel 4: out = (Wo(32x32) @ M_T(32x448)) * M_S  per (b,t).  WMMA f32.
// M_T is pixel-major, so each lane's B-fragment pair (c0,c0+1) is one b64.
// Output keeps the reference (B,T,32,H,W) layout.
// jobs = 64 * 2 mtiles * 28 ntiles = 3584; 8 waves/block.
// ---------------------------------------------------------------------------
__global__ void out_proj_kernel(const float* __restrict__ mtbuf,
                                const float* __restrict__ Wo,
                                const float* __restrict__ invz,
                                float* __restrict__ out) {
    const int lane = threadIdx.x & 31;
    const int wave = threadIdx.x >> 5;
    int job = blockIdx.x * 8 + wave;
    const int ntile = job % 28; job /= 28;
    const int mtile = job % 2;  job /= 2;
    const int bt    = job;                      // 0..63

    const int m     = lane & 15;
    const int khalf = lane >> 4;
    const int col   = ntile * 16 + m;
    const float* inp = mtbuf + ((size_t)bt * kHW + col) * kC;  // [32ch] row

    v8f acc = {0.f, 0.f, 0.f, 0.f, 0.f, 0.f, 0.f, 0.f};
#pragma unroll
    for (int k0 = 0; k0 < kC; k0 += 4) {
        const int c0 = k0 + 2 * khalf;
        v2f a, b;
        a[0] = Wo[(mtile * 16 + m) * kC + c0];
        a[1] = Wo[(mtile * 16 + m) * kC + c0 + 1];
        float2 bv = reinterpret_cast<const float2*>(inp + c0)[0]; // b64 load
        b[0] = bv.x;
        b[1] = bv.y;
        acc = wmma_f32_16x16x4(a, b, acc);
    }

    // M_S = max over heads of (1/Z) at this pixel column
    const float* iz = invz + (size_t)bt * kNH * kHW + col;
    float ms = iz[0];
    ms = fmaxf(ms, iz[kHW]);
    ms = fmaxf(ms, iz[2 * kHW]);
    ms = fmaxf(ms, iz[3 * kHW]);

    float* ob = out + (size_t)bt * kC * kHW;    // reference layout [c][p]
#pragma unroll
    for (int r = 0; r < 8; ++r) {
        const int row = mtile * 16 + r + 8 * khalf;
        ob[(size_t)row * kHW + col] = acc[r] * ms;
    }
}

// ---------------------------------------------------------------------------
extern "C" void kernel_launch(void* const* d_in, const int* in_sizes, int n_in,
                              void* d_out, int out_size, void* d_ws, size_t ws_size,
                              hipStream_t stream) {
    const float* first = (const float*)d_in[0];
    const float* x     = (const float*)d_in[1];
    const float* Wq    = (const float*)d_in[2];
    const float* Wk    = (const float*)d_in[3];
    const float* Wvp   = (const float*)d_in[4];
    const float* Wo    = (const float*)d_in[5];
    const float* gam   = (const float*)d_in[6];
    const float* bet   = (const float*)d_in[7];
    float* out = (float*)d_out;

    float* ws = (float*)d_ws;
    const size_t NTOT = (size_t)kB * kT * kC * kHW;   // 917504 floats
    float* qbuf  = ws;
    float* kbuf  = qbuf + NTOT;
    float* vbuf  = kbuf + NTOT;
    float* mtbuf = vbuf + NTOT;
    float* invz  = mtbuf + NTOT;                      // + B*T*NH*HW floats

    // 64*3*2*28 = 10752 wave-jobs / 8 waves = 1344 blocks
    proj_qkv_kernel<<<1344, 256, 0, stream>>>(first, x, Wq, Wk, Wvp,
                                              qbuf, kbuf, vbuf);
    // one block per (b,t,head), one thread per pixel
    ln_q_kernel<<<kB * kT * kNH, kHW, 0, stream>>>(qbuf, gam, bet);
    // one block per (b,head,t), one thread per pixel
    attn_kernel<<<kB * kNH * kT, kHW, 0, stream>>>(qbuf, kbuf, vbuf,
                                                   mtbuf, invz);
    // 64*2*28 = 3584 wave-jobs / 8 waves = 448 blocks
    out_proj_kernel<<<448, 256, 0, stream>>>(mtbuf, Wo, invz, out);
}